// DiffKF_88974542504593
// MI455X (gfx1250) — compile-verified
//
#include <hip/hip_runtime.h>
#include <hip/hip_bf16.h>

// ---------------------------------------------------------------------------
// DiffKF on MI455X (gfx1250, wave32).
// Key algebra: Kg = P_pred * inv(S) with S = P_pred + r*I  =>  Kg = I - r*inv(S)
//   P_upd = r*I - r^2 * inv(S)
//   m_upd = m_pred + d - r * inv(S) * d
// so the only heavy op is 64 SPD 360x360 inversions, done by Newton-Schulz
// (pure fp32 GEMMs -> v_wmma_f32_16x16x4_f32), padded to 384 with identity.
// ---------------------------------------------------------------------------

typedef __attribute__((ext_vector_type(2))) float v2f;
typedef __attribute__((ext_vector_type(8))) float v8f;

#define Dm   360
#define DP   384          // padded (24 tiles of 16)
#define MS   (DP * DP)    // floats per matrix
#define NTASK 64          // B*A = 2*32
#define NS_ITERS 10

__device__ __forceinline__ int srcmap(int g) {
    int o = g % 60;                 // block size 2*fut_len = 60
    return (o < 58) ? g + 2 : g;    // shift-by-one-step; last two slots repeat
}

// ---- build padded S = A P A^T + (1 + r_obs) on diag (gather form) ----------
__global__ void prep_S(const float* __restrict__ P, const float* __restrict__ rs,
                       float* __restrict__ wsmat, size_t slotStride, size_t offS,
                       int chunkStart, int nTasks) {
    int task = blockIdx.x / 576;                 // 576 blocks * 256 thr = 384*384
    if (task >= nTasks) return;
    int idx = (blockIdx.x % 576) * 256 + threadIdx.x;
    int i = idx / DP, j = idx % DP;
    int gt = chunkStart + task;
    float r_obs = rs[gt] + 1.0f;                 // OBSERVATION_INIT = 1.0
    float val;
    if (i < Dm && j < Dm) {
        int si = srcmap(i), sj = srcmap(j);
        val = P[(size_t)gt * Dm * Dm + (size_t)si * Dm + sj];
        if (i == j) val += 1.0f + r_obs;         // PROCESS_INIT + r_obs
    } else {
        val = (i == j) ? 1.0f : 0.0f;            // identity padding block
    }
    wsmat[(size_t)task * slotStride + offS + (size_t)i * DP + j] = val;
}

// ---- alpha = 1 / ||S||_inf per task (guarantees NS convergence) ------------
__global__ void row_norm(const float* __restrict__ wsmat, size_t slotStride,
                         size_t offS, float* __restrict__ alphas) {
    __shared__ float red[256];
    int task = blockIdx.x;
    const float* S = wsmat + (size_t)task * slotStride + offS;
    float mx = 0.0f;
    for (int row = threadIdx.x; row < DP; row += 256) {
        const float* r = S + (size_t)row * DP;
        float s = 0.0f;
        for (int j = 0; j < DP; ++j) s += fabsf(r[j]);
        mx = fmaxf(mx, s);
    }
    red[threadIdx.x] = mx;
    __syncthreads();
    for (int w = 128; w > 0; w >>= 1) {
        if (threadIdx.x < w) red[threadIdx.x] = fmaxf(red[threadIdx.x], red[threadIdx.x + w]);
        __syncthreads();
    }
    if (threadIdx.x == 0) alphas[task] = 1.0f / red[0];
}

// ---- X0 = alpha * I --------------------------------------------------------
__global__ void init_X(float* __restrict__ wsmat, size_t slotStride, size_t offX,
                       const float* __restrict__ alphas, int nTasks) {
    int task = blockIdx.x / 576;
    if (task >= nTasks) return;
    int idx = (blockIdx.x % 576) * 256 + threadIdx.x;
    int i = idx / DP, j = idx % DP;
    wsmat[(size_t)task * slotStride + offX + (size_t)i * DP + j] =
        (i == j) ? alphas[task] : 0.0f;
}

// ---- fp32 WMMA GEMM: mode 0: C = A*B ; mode 1: C = 2*A(at out) - A*B -------
// Each wave owns a 32x32 macro-tile (2x2 WMMA 16x16 tiles), K stepped by 4.
// A-frag (16x4): lane l<16 -> (M=l, K=k..k+1), lane l+16 -> (M=l, K=k+2..k+3)
// B-frag (4x16): lane l<16 -> (K=k..k+1, N=l), lane l+16 -> (K=k+2..k+3, N=l)
// C tile (16x16): vgpr v, lanes 0-15 -> (M=v, N=l); lanes 16-31 -> (M=v+8, N=l)
__global__ void gemm384(const float* __restrict__ wsmat_c, float* __restrict__ wsmat,
                        size_t slotStride, size_t offA, size_t offB, size_t offC,
                        int nTasks, int mode) {
    size_t gw = (size_t)blockIdx.x * 4 + (threadIdx.x >> 5);   // wave32
    int lane = threadIdx.x & 31;
    int task = (int)(gw / 144);                                 // 12x12 macro tiles
    if (task >= nTasks) return;
    int rem = (int)(gw % 144);
    int mt = rem / 12, nt = rem % 12;

    const float* Ap = wsmat_c + (size_t)task * slotStride + offA;
    const float* Bp = wsmat_c + (size_t)task * slotStride + offB;
    float*       Cp = wsmat   + (size_t)task * slotStride + offC;

    int half = lane >> 4;       // 0: K pair {k,k+1}, 1: {k+2,k+3}
    int l    = lane & 15;
    int koff = 2 * half;

    v8f acc[2][2];
    acc[0][0] = (v8f)0.0f; acc[0][1] = (v8f)0.0f;
    acc[1][0] = (v8f)0.0f; acc[1][1] = (v8f)0.0f;

    for (int k = 0; k < DP; k += 4) {
        v2f afr[2], bfr[2];
#pragma unroll
        for (int s = 0; s < 2; ++s) {
            int row = mt * 32 + s * 16 + l;
            afr[s] = *(const v2f*)(Ap + (size_t)row * DP + k + koff);   // contiguous pair
            int col = nt * 32 + s * 16 + l;
            bfr[s].x = Bp[(size_t)(k + koff)     * DP + col];
            bfr[s].y = Bp[(size_t)(k + koff + 1) * DP + col];
        }
#pragma unroll
        for (int si = 0; si < 2; ++si)
#pragma unroll
            for (int sj = 0; sj < 2; ++sj)
                acc[si][sj] = __builtin_amdgcn_wmma_f32_16x16x4_f32(
                    false, afr[si], false, bfr[sj], (short)0, acc[si][sj], false, false);
    }

#pragma unroll
    for (int si = 0; si < 2; ++si)
#pragma unroll
        for (int sj = 0; sj < 2; ++sj)
#pragma unroll
            for (int v = 0; v < 8; ++v) {
                int row = mt * 32 + si * 16 + v + 8 * half;
                int col = nt * 32 + sj * 16 + l;
                float val = acc[si][sj][v];
                if (mode) val = 2.0f * Ap[(size_t)row * DP + col] - val;  // X' = 2X - X*T
                Cp[(size_t)row * DP + col] = val;
            }
}

// ---- mean + initialized flag merge; matvec Sinv*d --------------------------
__global__ void finalize_mean(const float* __restrict__ mean, const float* __restrict__ obs,
                              const float* __restrict__ rs, const int* __restrict__ initialized,
                              const int* __restrict__ selected,
                              const float* __restrict__ wsmat, size_t slotStride, size_t offX,
                              int chunkStart, float* __restrict__ out_mean,
                              float* __restrict__ out_flag) {
    __shared__ float dS[Dm];
    __shared__ float mpS[Dm];
    int task = blockIdx.x;
    int gt = chunkStart + task;
    int i = threadIdx.x;
    bool sel = selected[gt] != 0;
    bool ini = initialized[gt] != 0;
    float r_obs = rs[gt] + 1.0f;
    const float* Sinv = wsmat + (size_t)task * slotStride + offX;

    if (i < Dm) {
        float mp = mean[(size_t)gt * Dm + srcmap(i)];
        mpS[i] = mp;
        dS[i]  = obs[(size_t)gt * Dm + i] - mp;
    }
    __syncthreads();
    if (i < Dm) {
        float out;
        if (sel && !ini) {
            out = obs[(size_t)gt * Dm + i];                    // init branch
        } else if (sel && ini) {
            const float* row = Sinv + (size_t)i * DP;
            float acc = 0.0f;
            for (int j = 0; j < Dm; ++j) acc = fmaf(row[j], dS[j], acc);
            out = mpS[i] + dS[i] - r_obs * acc;                // m_pred + (I - r*Sinv) d
        } else {
            out = mean[(size_t)gt * Dm + i];                   // passthrough
        }
        out_mean[(size_t)gt * Dm + i] = out;
    }
    if (i == 0) out_flag[gt] = (ini || sel) ? 1.0f : 0.0f;
}

// ---- covariance merge: upd -> r*I - r^2*Sinv ; init -> P + r*I ; else P ----
__global__ void finalize_cov(const float* __restrict__ P, const float* __restrict__ rs,
                             const int* __restrict__ initialized, const int* __restrict__ selected,
                             const float* __restrict__ wsmat, size_t slotStride, size_t offX,
                             int chunkStart, float* __restrict__ out_cov) {
    const int perTaskBlocks = 507;                // 507*256 >= 360*360
    int task = blockIdx.x / perTaskBlocks;
    int idx  = (blockIdx.x % perTaskBlocks) * 256 + threadIdx.x;
    if (idx >= Dm * Dm) return;
    int gt = chunkStart + task;
    int i = idx / Dm, j = idx % Dm;
    bool sel = selected[gt] != 0;
    bool ini = initialized[gt] != 0;
    float r_obs = rs[gt] + 1.0f;
    float val;
    if (sel && ini) {
        float sinv = wsmat[(size_t)task * slotStride + offX + (size_t)i * DP + j];
        val = ((i == j) ? r_obs : 0.0f) - r_obs * r_obs * sinv;
    } else {
        val = P[(size_t)gt * Dm * Dm + idx];
        if (sel && !ini && i == j) val += r_obs;
    }
    out_cov[(size_t)gt * Dm * Dm + idx] = val;
}

// ---------------------------------------------------------------------------
extern "C" void kernel_launch(void* const* d_in, const int* in_sizes, int n_in,
                              void* d_out, int out_size, void* d_ws, size_t ws_size,
                              hipStream_t stream) {
    const float* mean = (const float*)d_in[0];
    const float* cov  = (const float*)d_in[1];
    const float* obs  = (const float*)d_in[2];
    const float* rs   = (const float*)d_in[3];
    const int*   initialized = (const int*)d_in[4];
    const int*   selected    = (const int*)d_in[5];
    // d_in[6]=K, d_in[7]=fut_len: compile-time constants (6, 30) here.

    float* out_mean = (float*)d_out;
    float* out_cov  = out_mean + NTASK * Dm;            // 23040
    float* out_flag = out_cov  + (size_t)NTASK * Dm * Dm;

    // workspace layout: [alphas: 1024B pad][slots: 4 matrices of 384x384 f32 each]
    float*  alphas = (float*)d_ws;
    float*  wsmat  = (float*)((char*)d_ws + 1024);
    const size_t slotStride = (size_t)4 * MS;           // floats per slot
    const size_t slotBytes  = slotStride * sizeof(float);
    size_t avail = ws_size > 1024 ? ws_size - 1024 : 0;
    int slots = (int)(avail / slotBytes);
    if (slots < 1)  slots = 1;                          // best effort
    if (slots > NTASK) slots = NTASK;

    const size_t offS = 0, offT = MS, offX0 = 2 * (size_t)MS, offX1 = 3 * (size_t)MS;

    for (int chunkStart = 0; chunkStart < NTASK; chunkStart += slots) {
        int n = NTASK - chunkStart;
        if (n > slots) n = slots;

        prep_S  <<<n * 576, 256, 0, stream>>>(cov, rs, wsmat, slotStride, offS, chunkStart, n);
        row_norm<<<n,       256, 0, stream>>>(wsmat, slotStride, offS, alphas);
        init_X  <<<n * 576, 256, 0, stream>>>(wsmat, slotStride, offX0, alphas, n);

        size_t oX = offX0, oXn = offX1;
        for (int it = 0; it < NS_ITERS; ++it) {
            // T = S * X
            gemm384<<<n * 36, 128, 0, stream>>>(wsmat, wsmat, slotStride, offS, oX, offT, n, 0);
            // X' = 2*X - X * T   (Newton-Schulz)
            gemm384<<<n * 36, 128, 0, stream>>>(wsmat, wsmat, slotStride, oX, offT, oXn, n, 1);
            size_t t = oX; oX = oXn; oXn = t;
        }

        finalize_mean<<<n, 384, 0, stream>>>(mean, obs, rs, initialized, selected,
                                             wsmat, slotStride, oX, chunkStart,
                                             out_mean, out_flag);
        finalize_cov<<<n * 507, 256, 0, stream>>>(cov, rs, initialized, selected,
                                                  wsmat, slotStride, oX, chunkStart, out_cov);
    }
}